// CRFLayer_4544075399226
// MI455X (gfx1250) — compile-verified
//
#include <hip/hip_runtime.h>
#include <math.h>

typedef __attribute__((ext_vector_type(2))) float v2f;
typedef __attribute__((ext_vector_type(8))) float v8f;

#define TAGS   64
#define TILE_B 16   // batch rows per workgroup (= WMMA M)
#define NTHREADS 128 // 4 waves * 32 lanes; 4 N-tiles of 16 cover TAGS=64

// ---------------------------------------------------------------------------
// Kernel 1: gold path score per batch.
// gold[b] = sum_s em[b,s,lab]*mask + sum_s T[lab_s,lab_{s+1}]*mask_s*mask_{s+1}
// ---------------------------------------------------------------------------
__global__ void crf_gold_kernel(const float* __restrict__ feat,
                                const int*   __restrict__ labels,
                                const float* __restrict__ mask,
                                const float* __restrict__ trans,
                                float* __restrict__ gold, int S) {
  const int b   = blockIdx.x;
  const int tid = threadIdx.x;          // 64 threads
  __shared__ float red[64];
  float acc = 0.f;
  for (int s = tid; s < S; s += 64) {
    const int lab = labels[b * S + s];
    acc += feat[((size_t)b * S + s) * TAGS + lab] * mask[b * S + s];
    if (s + 1 < S) {
      const int lab2 = labels[b * S + s + 1];
      acc += trans[lab * TAGS + lab2] * mask[b * S + s] * mask[b * S + s + 1];
    }
  }
  red[tid] = acc;
  __syncthreads();
  for (int off = 32; off > 0; off >>= 1) {
    if (tid < off) red[tid] += red[tid + off];
    __syncthreads();
  }
  if (tid == 0) gold[b] = red[0];
}

// ---------------------------------------------------------------------------
// Kernel 2: log-partition forward scan in exp-space via fp32 WMMA.
// Per step:  Q(16x64) = P(16x64) x E(64x64)  with 16 chained
// v_wmma_f32_16x16x4_f32 per wave (each wave owns a 16-col N tile).
// alpha'[r,c] = log(max(Q,1e-30)) + m[r] + em[b,t,c], mask-blended.
// ---------------------------------------------------------------------------
__global__ void crf_logz_kernel(const float* __restrict__ feat,
                                const float* __restrict__ mask,
                                const float* __restrict__ trans,
                                float* __restrict__ logZ, int S) {
  __shared__ float sAlpha[TILE_B][TAGS];
  __shared__ float sP[TILE_B][TAGS];
  __shared__ float sM[TILE_B];

  const int tid     = threadIdx.x;      // 0..127
  const int wave    = tid >> 5;         // 0..3  -> N tile [16*wave, 16*wave+16)
  const int lane    = tid & 31;
  const int half    = lane >> 4;        // 0: lanes 0-15, 1: lanes 16-31
  const int l16     = lane & 15;
  const int b0      = blockIdx.x * TILE_B;
  const int colBase = wave * 16;
  const int myCol   = colBase + l16;

  // B fragments: E = exp(transitions), resident in VGPRs for the whole scan.
  // B(4x16) layout: lanes0-15: v0=K0,v1=K1 ; lanes16-31: v0=K2,v1=K3.
  v2f Bfrag[16];
#pragma unroll
  for (int kc = 0; kc < 16; ++kc) {
    const int k0 = kc * 4 + half * 2;
    Bfrag[kc].x = expf(trans[(k0 + 0) * TAGS + myCol]);
    Bfrag[kc].y = expf(trans[(k0 + 1) * TAGS + myCol]);
  }

  // alpha0 = emissions at t=0
  for (int idx = tid; idx < TILE_B * TAGS; idx += NTHREADS) {
    const int r = idx >> 6, c = idx & 63;
    sAlpha[r][c] = feat[((size_t)(b0 + r) * S) * TAGS + c];
  }
  __syncthreads();
  if (tid < TILE_B) {
    float mx = -3.0e38f;
    for (int c = 0; c < TAGS; ++c) mx = fmaxf(mx, sAlpha[tid][c]);
    sM[tid] = mx;
  }
  __syncthreads();
  for (int idx = tid; idx < TILE_B * TAGS; idx += NTHREADS) {
    const int r = idx >> 6, c = idx & 63;
    sP[r][c] = expf(sAlpha[r][c] - sM[r]);
  }
  __syncthreads();

  for (int t = 1; t < S; ++t) {
    // Q = P x E for this wave's 16-column tile, K=64 in 16 chunks of 4.
    v8f acc = {};
#pragma unroll
    for (int kc = 0; kc < 16; ++kc) {
      // A(16x4) layout: lane M=l16; lanes0-15 carry K0,K1; lanes16-31 K2,K3.
      const int k0 = kc * 4 + half * 2;
      v2f a;
      a.x = sP[l16][k0];
      a.y = sP[l16][k0 + 1];
      acc = __builtin_amdgcn_wmma_f32_16x16x4_f32(
          /*neg_a=*/false, a, /*neg_b=*/false, Bfrag[kc],
          /*c_mod=*/(short)0, acc, /*reuse_a=*/false, /*reuse_b=*/false);
    }
    // C/D layout: vgpr k, lanes0-15 -> row k, lanes16-31 -> row k+8; col=lane%16.
#pragma unroll
    for (int k = 0; k < 8; ++k) {
      const int r = k + half * 8;
      const int b = b0 + r;
      const float q  = fmaxf(acc[k], 1e-30f);
      const float em = feat[((size_t)b * S + t) * TAGS + myCol];
      const float an = logf(q) + sM[r] + em;
      const float mt = mask[(size_t)b * S + t];
      sAlpha[r][myCol] = an * mt + sAlpha[r][myCol] * (1.f - mt);
      if (t + 1 < S)
        __builtin_prefetch(&feat[((size_t)b * S + t + 1) * TAGS + myCol], 0, 1);
    }
    __syncthreads();
    if (tid < TILE_B) {
      float mx = -3.0e38f;
      for (int c = 0; c < TAGS; ++c) mx = fmaxf(mx, sAlpha[tid][c]);
      sM[tid] = mx;
    }
    __syncthreads();
    for (int idx = tid; idx < TILE_B * TAGS; idx += NTHREADS) {
      const int r = idx >> 6, c = idx & 63;
      sP[r][c] = expf(sAlpha[r][c] - sM[r]);
    }
    __syncthreads();
  }

  // logZ[b] = m[b] + log(sum_j exp(alpha - m)) = m[b] + log(sum_j P)
  if (tid < TILE_B) {
    float s = 0.f;
    for (int c = 0; c < TAGS; ++c) s += sP[tid][c];
    logZ[b0 + tid] = sM[tid] + logf(s);
  }
}

// ---------------------------------------------------------------------------
// Kernel 3: out = mean(logZ - gold)
// ---------------------------------------------------------------------------
__global__ void crf_reduce_kernel(const float* __restrict__ logZ,
                                  const float* __restrict__ gold,
                                  float* __restrict__ out, int B) {
  __shared__ float red[NTHREADS];
  const int tid = threadIdx.x;
  float v = 0.f;
  for (int b = tid; b < B; b += NTHREADS) v += logZ[b] - gold[b];
  red[tid] = v;
  __syncthreads();
  for (int off = NTHREADS / 2; off > 0; off >>= 1) {
    if (tid < off) red[tid] += red[tid + off];
    __syncthreads();
  }
  if (tid == 0) out[0] = red[0] / (float)B;
}

extern "C" void kernel_launch(void* const* d_in, const int* in_sizes, int n_in,
                              void* d_out, int out_size, void* d_ws, size_t ws_size,
                              hipStream_t stream) {
  const float* feat   = (const float*)d_in[0];
  const int*   labels = (const int*)d_in[1];
  const float* mask   = (const float*)d_in[2];
  const float* trans  = (const float*)d_in[3];

  const int S  = 512;
  const int BS = in_sizes[2];
  const int B  = BS / S; // 128

  float* gold = (float*)d_ws;
  float* logZ = gold + B;

  crf_gold_kernel<<<B, 64, 0, stream>>>(feat, labels, mask, trans, gold, S);
  crf_logz_kernel<<<B / TILE_B, NTHREADS, 0, stream>>>(feat, mask, trans, logZ, S);
  crf_reduce_kernel<<<1, NTHREADS, 0, stream>>>(logZ, gold, (float*)d_out, B);
}